// GenerativeMPSBase_58574763983327
// MI455X (gfx1250) — compile-verified
//
#include <hip/hip_runtime.h>

// Fused MPS contraction for gfx1250 (MI455X), f32 WMMA 16x16x4.
// Block 0: norm_sq transfer-matrix chain. Blocks 1..8: batched forward chain
// (32 batches each) with TDM (tensor_load_to_lds) double-buffered slab staging.
// ws: [0]=norm_sq (f32), [1]=ready flag (u32).

#define N_SITES 784
#define BD      128
#define BATCH   256
#define TB      32      // batches per forward block
#define FP      132     // forward Al pitch (dwords)
#define AP      132     // norm AlL pitch
#define TP      130     // norm Ttr pitch (2*130 % 64 == 4 -> conflict-free f2 reads)
#define PB      272     // staged slab row pitch (dwords): 2*272 % 64 == 32
#define SLABF   (BD * PB)                 // floats per staged slab buffer
#define SMEMB   (2 * SLABF * 4)           // dynamic LDS bytes (278528)
#define PI_2    1.57079632679489661923f

typedef __attribute__((ext_vector_type(2))) float        v2f;
typedef __attribute__((ext_vector_type(8))) float        v8f;
typedef __attribute__((ext_vector_type(4))) int          v4i;
typedef __attribute__((ext_vector_type(8))) int          v8i;
typedef __attribute__((ext_vector_type(4))) unsigned int v4u;
typedef __attribute__((address_space(1))) v4i*           gv4p;
typedef __attribute__((address_space(3))) v4i*           lv4p;

#if __has_builtin(__builtin_amdgcn_tensor_load_to_lds) && \
    __has_builtin(__builtin_amdgcn_s_wait_tensorcnt)
#define HAVE_TDM 1
#else
#define HAVE_TDM 0
#endif

__device__ __forceinline__ v8f wmma_f32(v2f a, v2f b, v8f c) {
  return __builtin_amdgcn_wmma_f32_16x16x4_f32(false, a, false, b, (short)0, c,
                                               false, false);
}

#if HAVE_TDM
// One TDM op copies a whole 128x256-elem f32 slab into LDS, inserting a
// 16-dword pad after every 256 dwords (row) -> LDS row pitch PB=272 dwords.
__device__ __forceinline__ void tdm_load_slab(const float* gsrc, float* lds_dst) {
  const unsigned long long ga = (unsigned long long)(uintptr_t)gsrc;
  v4u g0;
  g0[0] = 1u;                                     // count=1, user descriptor
  g0[1] = (unsigned)(uintptr_t)lds_dst;           // lds_addr [63:32]
  g0[2] = (unsigned)ga;                           // global_addr[31:0]
  g0[3] = ((unsigned)(ga >> 32) & 0x01FFFFFFu)    // global_addr[56:32]
          | (2u << 30);                           // type=2 ("image")
  v8i g1;
  g1[0] = (int)((2u << 16)        // data_size=4B
              | (1u << 20)        // pad_enable
              | (7u << 22)        // pad_interval: every 256 dwords
              | (15u << 25));     // pad_amount: 16 dwords
  g1[1] = (int)(256u << 16);      // tensor_dim0 = 256  (bits 79:48)
  g1[2] = (int)(128u << 16);      // tensor_dim1 = 128  (bits 111:80)
  g1[3] = (int)(256u << 16);      // tile_dim0   = 256  (bits 127:112)
  g1[4] = (int)128;               // tile_dim1   = 128  (bits 143:128)
  g1[5] = (int)256;               // tensor_dim0_stride = 256 (bits 207:160)
  g1[6] = 0;
  g1[7] = 0;
  const v4i gz4 = {0, 0, 0, 0};   // 2D tensor: groups 2/3 unused
  const v8i gz8 = {0, 0, 0, 0, 0, 0, 0, 0};
  __builtin_amdgcn_tensor_load_to_lds(g0, g1, gz4, gz4, gz8, 0);
}
__device__ __forceinline__ void stage_wait() {
  __builtin_amdgcn_s_wait_tensorcnt(0);           // no-op for non-issuing waves
}
#else
__device__ __forceinline__ void async_cp16(float* lds_dst, const float* gsrc) {
#if __has_builtin(__builtin_amdgcn_global_load_async_to_lds_b128)
  __builtin_amdgcn_global_load_async_to_lds_b128(
      (gv4p)(uintptr_t)gsrc, (lv4p)(uintptr_t)lds_dst, 0, 0);
#else
  asm volatile("global_load_async_to_lds_b128 %0, %1, off"
               :: "v"((unsigned)(uintptr_t)lds_dst),
                  "v"((unsigned long long)(uintptr_t)gsrc)
               : "memory");
#endif
}
__device__ __forceinline__ void stage_wait() {
#if __has_builtin(__builtin_amdgcn_s_wait_asynccnt)
  __builtin_amdgcn_s_wait_asynccnt(0);
#else
  asm volatile("s_wait_asynccnt 0" ::: "memory");
#endif
}
#endif

// issue staging of one slab into dst (TDM: wave 0 only; async: all threads)
__device__ __forceinline__ void stage_issue(const float* src, float* dst,
                                            int tid, int wave) {
#if HAVE_TDM
  (void)tid;
  if (wave == 0) tdm_load_slab(src, dst);
#else
  (void)wave;
#pragma unroll
  for (int i = 0; i < 8; ++i) {
    const int g = tid + (i << 10);                // 16B chunk id, 8192 total
    async_cp16(dst + (g >> 6) * PB + (g & 63) * 4, src + g * 4);
  }
#endif
}

__global__ void mps_reset(unsigned* ws) {
  if (threadIdx.x == 0) ws[1] = 0u;               // clear ready flag
}

__global__ __launch_bounds__(1024, 1) void mps_fused(
    const float* __restrict__ x, const float* __restrict__ M,
    float* __restrict__ ws, float* __restrict__ out)
{
  extern __shared__ float sm[];
  __shared__ float  Al[2][TB][FP];
  __shared__ float2 eb[2][TB];
  __shared__ float  red[TB][32];

  const int tid  = threadIdx.x;
  const int lane = tid & 31;
  const int wave = tid >> 5;            // 0..31
  const int mr   = lane & 15;
  const int khi  = (lane >> 4) << 1;    // K offset +0 / +2 per half-wave
  const int hi8  = (lane >> 4) << 3;    // C/D row offset +0 / +8

  if (blockIdx.x == 0) {
    // =================== norm_sq chain (one block, 32 waves) ===============
    float* AlL = sm;                    // [BD][AP]   AlL[b][a] = Al[a][b]
    float* Ttr = sm + BD * AP;          // [256][TP]  Ttr[n][b] = T[b][n]

    for (int idx = tid; idx < BD * BD; idx += 1024) {
      const int a = idx >> 7, b = idx & (BD - 1);
      AlL[b * AP + a] = M[2 * a] * M[2 * b] + M[2 * a + 1] * M[2 * b + 1];
    }
    __syncthreads();

    const int mt = wave >> 2;           // G1 b-tile / G2 r-tile (0..7)
    const int nb = (wave & 3) * 64;     // G1 col base (4 tiles)
    const int sb = (wave & 3) * 2;      // G2 s-tile base (2 tiles)

    for (int t = 1; t <= N_SITES - 2; ++t) {
      const float* slab = M + (size_t)t * (BD * BD * 2);
      if (t < N_SITES - 2)              // pull next slab toward L2/L0
        __builtin_prefetch(slab + BD * BD * 2 + tid * 32, 0, 0);

      // G1: T[b,n] = sum_a Al[a,b]*Mflat[a,n];   4 tiles share the A fragment
      v8f acc[4] = {};
      for (int k0 = 0; k0 < BD; k0 += 4) {
        const int ka = k0 + khi;
        const v2f av = *(const v2f*)&AlL[(mt * 16 + mr) * AP + ka];
        const float* r0 = slab + (size_t)ka * 256 + nb + mr;
#pragma unroll
        for (int q = 0; q < 4; ++q) {
          v2f bv; bv.x = r0[q * 16]; bv.y = r0[256 + q * 16];
          acc[q] = wmma_f32(av, bv, acc[q]);
        }
      }
#pragma unroll
      for (int q = 0; q < 4; ++q)
        for (int j = 0; j < 8; ++j)
          Ttr[(nb + q * 16 + mr) * TP + mt * 16 + j + hi8] = acc[q][j];
      __syncthreads();

      // G2: new[r,s] = sum_b Ttr[2r][b]*Mf[b,2s] + Ttr[2r+1][b]*Mf[b,2s+1]
      const int rr = mt * 16 + mr;
      v8f acc2[2] = {};
      for (int k0 = 0; k0 < BD; k0 += 4) {
        const int ka = k0 + khi;
        const v2f a0 = *(const v2f*)&Ttr[(2 * rr + 0) * TP + ka];
        const v2f a1 = *(const v2f*)&Ttr[(2 * rr + 1) * TP + ka];
        const float* r0 = slab + (size_t)ka * 256;
#pragma unroll
        for (int q = 0; q < 2; ++q) {
          const int ss = (sb + q) * 16 + mr;
          const float2 g0 = *(const float2*)(r0 + 2 * ss);
          const float2 g1 = *(const float2*)(r0 + 256 + 2 * ss);
          v2f b0; b0.x = g0.x; b0.y = g1.x;
          v2f b1; b1.x = g0.y; b1.y = g1.y;
          acc2[q] = wmma_f32(a0, b0, acc2[q]);
          acc2[q] = wmma_f32(a1, b1, acc2[q]);
        }
      }
#pragma unroll
      for (int q = 0; q < 2; ++q)
        for (int j = 0; j < 8; ++j)     // store transposed: AlL[s][r]
          AlL[((sb + q) * 16 + mr) * AP + mt * 16 + j + hi8] = acc2[q][j];
      __syncthreads();
    }

    // norm = sum_{a,b} Al[a,b] * (An[a,:].An[b,:]),  An = MPS[783,:,0,:]
    float part = 0.f;
    for (int idx = tid; idx < BD * BD; idx += 1024) {
      const int a = idx >> 7, b = idx & (BD - 1);
      const size_t oa = ((size_t)(N_SITES - 1) * BD + a) * (BD * 2);
      const size_t ob = ((size_t)(N_SITES - 1) * BD + b) * (BD * 2);
      part += AlL[b * AP + a] * (M[oa] * M[ob] + M[oa + 1] * M[ob + 1]);
    }
    float* r = Ttr;
    r[tid] = part;
    __syncthreads();
    for (int s = 512; s > 0; s >>= 1) {
      if (tid < s) r[tid] += r[tid + s];
      __syncthreads();
    }
    if (tid == 0) {
      *(volatile float*)&ws[0] = r[0];
      __threadfence();                  // release norm before flag
      *(volatile unsigned*)&((unsigned*)ws)[1] = 1u;
    }
    return;
  }

  // ===================== forward chain (blocks 1..8) =======================
  float* slab0 = sm;                    // staged slab, double-buffered
  float* slab1 = sm + SLABF;
  const int b0 = ((int)blockIdx.x - 1) * TB;
  const int mt = wave >> 3;             // 0..1 (waves 0..15 compute)
  const int nt = (wave & 7) * 16;

  // kick off staging of site-1 slab while doing site-0 init
  stage_issue(M + (size_t)(BD * BD * 2), slab0, tid, wave);

  for (int idx = tid; idx < TB * BD; idx += 1024) {
    const int b = idx >> 7, l = idx & (BD - 1);
    const float xv = x[(b0 + b) * N_SITES];
    Al[0][b][l] = __cosf(PI_2 * xv) * M[2 * l] + __sinf(PI_2 * xv) * M[2 * l + 1];
  }

  for (int t = 1; t <= N_SITES - 2; ++t) {
    const int cur = (t - 1) & 1, nxt = t & 1;
    if (tid < TB) {
      const float xv = x[(b0 + tid) * N_SITES + t];
      eb[nxt][tid] = make_float2(__cosf(PI_2 * xv), __sinf(PI_2 * xv));
    }
    stage_wait();                       // slab for site t is now in LDS
    __syncthreads();

    if (t < N_SITES - 2)                // overlap: stage site t+1 slab
      stage_issue(M + (size_t)(t + 1) * (BD * BD * 2),
                  (t & 1) ? slab1 : slab0, tid, wave);

    if (wave < 16) {
      const float* bs = cur ? slab1 : slab0;
      v8f acc0 = {}, acc1 = {};
      for (int k0 = 0; k0 < BD; k0 += 4) {
        const int ka = k0 + khi;
        const v2f av = *(const v2f*)&Al[cur][mt * 16 + mr][ka];
        const float2 g0 = *(const float2*)(bs + (size_t)ka * PB + (nt + mr) * 2);
        const float2 g1 = *(const float2*)(bs + (size_t)(ka + 1) * PB + (nt + mr) * 2);
        v2f bv0; bv0.x = g0.x; bv0.y = g1.x;     // channel 0
        v2f bv1; bv1.x = g0.y; bv1.y = g1.y;     // channel 1
        acc0 = wmma_f32(av, bv0, acc0);
        acc1 = wmma_f32(av, bv1, acc1);
      }
#pragma unroll
      for (int j = 0; j < 8; ++j) {              // per-batch channel combine
        const int m = mt * 16 + j + hi8;
        const float2 e = eb[nxt][m];
        Al[nxt][m][nt + mr] = e.x * acc0[j] + e.y * acc1[j];
      }
    }
  }
  __syncthreads();

  // right boundary + per-batch dot
  const int fb = tid >> 5, fl = tid & 31;
  const float xv = x[(b0 + fb) * N_SITES + (N_SITES - 1)];
  const float c = __cosf(PI_2 * xv), s = __sinf(PI_2 * xv);
  float part = 0.f;
  for (int l = fl; l < BD; l += 32) {
    const size_t o = ((size_t)(N_SITES - 1) * BD + l) * (BD * 2);
    part += Al[0][fb][l] * (c * M[o] + s * M[o + 1]);
  }
  red[fb][fl] = part;
  __syncthreads();

  if (tid == 0) {                       // wait for norm_sq from block 0
    while (*(volatile unsigned*)&((unsigned*)ws)[1] != 1u)
      __builtin_amdgcn_s_sleep(16);
  }
  __syncthreads();
  __threadfence();
  if (tid < TB) {
    float d = 0.f;
    for (int i = 0; i < 32; ++i) d += red[tid][i];
    out[b0 + tid] = d * d / (*(volatile const float*)&ws[0]);
  }
}

extern "C" void kernel_launch(void* const* d_in, const int* in_sizes, int n_in,
                              void* d_out, int out_size, void* d_ws, size_t ws_size,
                              hipStream_t stream) {
  (void)in_sizes; (void)n_in; (void)out_size; (void)ws_size;
  const float* x   = (const float*)d_in[0];
  const float* MPS = (const float*)d_in[1];

  (void)hipFuncSetAttribute(reinterpret_cast<const void*>(mps_fused),
                            hipFuncAttributeMaxDynamicSharedMemorySize, SMEMB);

  mps_reset<<<1, 32, 0, stream>>>((unsigned*)d_ws);
  mps_fused<<<1 + BATCH / TB, 1024, SMEMB, stream>>>(x, MPS, (float*)d_ws,
                                                     (float*)d_out);
}